// CWE416_53618371723275
// MI455X (gfx1250) — compile-verified
//
#include <hip/hip_runtime.h>

#define IN_CH 116
#define HID   256
#define OUTC  2

typedef float v2f __attribute__((ext_vector_type(2)));
typedef float v8f __attribute__((ext_vector_type(8)));

// Non-returning hardware fp32 atomic add at device scope (L2 atomic units).
__device__ __forceinline__ void atomic_add_f32(float* p, float v) {
  asm volatile("global_atomic_add_f32 %0, %1, off scope:SCOPE_DEV"
               :: "v"(p), "v"(v)
               : "memory");
}

// ---------------- degree / normalization ----------------
__global__ void k_fill_ones(float* __restrict__ p, int n) {
  int i = blockIdx.x * blockDim.x + threadIdx.x;
  if (i < n) p[i] = 1.0f;  // self-loop contributes 1 to degree
}

__global__ void k_count_deg(const int* __restrict__ dst, float* __restrict__ deg, int E) {
  int i = blockIdx.x * blockDim.x + threadIdx.x;
  if (i < E) atomic_add_f32(&deg[dst[i]], 1.0f);
}

__global__ void k_rsqrt(float* __restrict__ p, int n) {
  int i = blockIdx.x * blockDim.x + threadIdx.x;
  if (i < n) p[i] = rsqrtf(p[i]);
}

// ---------------- fp32 WMMA GEMM: hw[N,M] = H[N,K] @ W[K,M] ----------------
// One wave per 16x16 output tile; K stepped by 4 via V_WMMA_F32_16X16X4_F32.
// Optional fused epilogue: agg[n,f] = hw[n,f] * dinv[n]^2 (self-loop init).
// Grid: (N/16, M/(16*8)), block = 256 threads = 8 waves.
__global__ __launch_bounds__(256) void k_gemm_wmma(const float* __restrict__ H,
                                                   const float* __restrict__ W,
                                                   float* __restrict__ hw,
                                                   float* __restrict__ agg,  // may be null
                                                   const float* __restrict__ dinv,
                                                   int K, int M) {
  const int wave = threadIdx.x >> 5;
  const int lane = threadIdx.x & 31;
  const int half = lane >> 4;   // 0: lanes 0-15, 1: lanes 16-31
  const int l    = lane & 15;
  const int row0 = blockIdx.x * 16;
  const int col  = (blockIdx.y * 8 + wave) * 16 + l;

  // A 16x4 fragment: lane holds A[row][k+2*half .. k+2*half+1] (8B-aligned pair)
  const float* ap  = H + (size_t)(row0 + l) * K + 2 * half;
  // B 4x16 fragment: v0 = B[k+2*half][col], v1 = B[k+2*half+1][col]
  const float* wp0 = W + (size_t)(2 * half) * M + col;
  const float* wp1 = wp0 + M;
  const size_t wstep = 4 * (size_t)M;

  v8f acc = {0.f, 0.f, 0.f, 0.f, 0.f, 0.f, 0.f, 0.f};
  for (int k = 0; k < K; k += 4) {
    v2f a = *(const v2f*)ap;
    v2f b;
    b.x = *wp0;
    b.y = *wp1;
    ap  += 4;
    wp0 += wstep;
    wp1 += wstep;
    acc = __builtin_amdgcn_wmma_f32_16x16x4_f32(
        /*neg_a=*/false, a, /*neg_b=*/false, b,
        /*c_mod=*/(short)0, acc, /*reuse_a=*/false, /*reuse_b=*/false);
  }

  // C/D 16x16 f32: VGPR v -> row row0 + v + 8*half, col
  const int rbase = row0 + 8 * half;
  float* oh = hw + (size_t)rbase * M + col;
#pragma unroll
  for (int v = 0; v < 8; ++v) oh[(size_t)v * M] = acc[v];

  if (agg != nullptr) {  // uniform branch
    float* oa = agg + (size_t)rbase * M + col;
#pragma unroll
    for (int v = 0; v < 8; ++v) {
      float d = dinv[rbase + v];
      oa[(size_t)v * M] = acc[v] * d * d;
    }
  }
}

// ---------------- tiny GEMM for the 2-wide output layer ----------------
// Fused: hw3 = h @ W3, out = hw3 * dinv^2 (self-loop init of final agg)
__global__ void k_gemm_small(const float* __restrict__ H, const float* __restrict__ W,
                             float* __restrict__ hw, float* __restrict__ out,
                             const float* __restrict__ dinv,
                             int K, int M, int total) {
  int i = blockIdx.x * blockDim.x + threadIdx.x;
  if (i >= total) return;
  int n = i / M, m = i - n * M;
  const float* h = H + (size_t)n * K;
  float s = 0.f;
  for (int k = 0; k < K; ++k) s = fmaf(h[k], W[(size_t)k * M + m], s);
  hw[i] = s;
  float d = dinv[n];
  out[i] = s * d * d;
}

// ---------------- GCN aggregation ----------------
// agg[n,f] = hw[n,f] * dinv[n]^2 (standalone; used where fusion would alias)
__global__ void k_scale_self(const float* __restrict__ hw, const float* __restrict__ dinv,
                             float* __restrict__ agg, int M, int total) {
  int i = blockIdx.x * blockDim.x + threadIdx.x;
  if (i >= total) return;
  int n = i / M;
  float d = dinv[n];
  agg[i] = hw[i] * d * d;
}

// agg[dst] += hw[src] * dinv[src]*dinv[dst], M == 256, 64 threads x float4 per edge
__global__ void k_edge_agg256(const int* __restrict__ src, const int* __restrict__ dst,
                              const float* __restrict__ dinv, const float* __restrict__ hw,
                              float* __restrict__ agg, int E) {
  long t = (long)blockIdx.x * blockDim.x + threadIdx.x;
  if (t >= (long)E * 64) return;
  int e = (int)(t >> 6);
  int f = (int)(t & 63) << 2;
  int s = src[e], d = dst[e];
  float c = dinv[s] * dinv[d];
  const float4 v = *(const float4*)(hw + (size_t)s * 256 + f);
  float* a = agg + (size_t)d * 256 + f;
  atomic_add_f32(a + 0, v.x * c);
  atomic_add_f32(a + 1, v.y * c);
  atomic_add_f32(a + 2, v.z * c);
  atomic_add_f32(a + 3, v.w * c);
}

// M == 2 variant: one thread per edge
__global__ void k_edge_agg2(const int* __restrict__ src, const int* __restrict__ dst,
                            const float* __restrict__ dinv, const float* __restrict__ hw,
                            float* __restrict__ agg, int E) {
  int e = blockIdx.x * blockDim.x + threadIdx.x;
  if (e >= E) return;
  int s = src[e], d = dst[e];
  float c = dinv[s] * dinv[d];
  atomic_add_f32(&agg[(size_t)d * 2 + 0], hw[(size_t)s * 2 + 0] * c);
  atomic_add_f32(&agg[(size_t)d * 2 + 1], hw[(size_t)s * 2 + 1] * c);
}

// h = relu(agg + b)      (M is a power of two: 256 or 2)
__global__ void k_bias_relu(const float* __restrict__ agg, const float* __restrict__ b,
                            float* __restrict__ h, int M, int total) {
  int i = blockIdx.x * blockDim.x + threadIdx.x;
  if (i >= total) return;
  float v = agg[i] + b[i & (M - 1)];
  h[i] = v > 0.f ? v : 0.f;
}

// out += b (final layer, no activation, in place)
__global__ void k_bias(float* __restrict__ out, const float* __restrict__ b,
                       int M, int total) {
  int i = blockIdx.x * blockDim.x + threadIdx.x;
  if (i >= total) return;
  out[i] += b[i & (M - 1)];
}

static inline int cdiv(long a, long b) { return (int)((a + b - 1) / b); }

extern "C" void kernel_launch(void* const* d_in, const int* in_sizes, int n_in,
                              void* d_out, int out_size, void* d_ws, size_t ws_size,
                              hipStream_t stream) {
  const float* x  = (const float*)d_in[0];
  const int*   ei = (const int*)d_in[1];
  const float* W1 = (const float*)d_in[2];
  const float* b1 = (const float*)d_in[3];
  const float* W2 = (const float*)d_in[4];
  const float* b2 = (const float*)d_in[5];
  const float* W3 = (const float*)d_in[6];
  const float* b3 = (const float*)d_in[7];
  float* out = (float*)d_out;

  const int N = in_sizes[0] / IN_CH;   // 50000 (multiple of 16)
  const int E = in_sizes[1] / 2;       // 1.6M
  const int* src = ei;
  const int* dst = ei + E;

  // workspace: dinv[N] | bufA[N*HID] | bufB[N*HID] | hw3[N*OUTC]
  float* dinv = (float*)d_ws;
  float* bufA = dinv + N;
  float* bufB = bufA + (size_t)N * HID;
  float* hw3  = bufB + (size_t)N * HID;

  const int TB = 256;
  const int NH = N * HID;
  const int NO = N * OUTC;

  // degree -> dinv = 1/sqrt(in-degree + 1)
  k_fill_ones<<<cdiv(N, TB), TB, 0, stream>>>(dinv, N);
  k_count_deg<<<cdiv(E, TB), TB, 0, stream>>>(dst, dinv, E);
  k_rsqrt<<<cdiv(N, TB), TB, 0, stream>>>(dinv, N);

  dim3 gGemm(N / 16, HID / (16 * 8));  // one wave per 16x16 tile

  // Layer 1 (fused agg init): hw1 -> bufA, agg -> bufB; h1 = relu(agg+b1) -> bufA
  k_gemm_wmma<<<gGemm, TB, 0, stream>>>(x, W1, bufA, bufB, dinv, IN_CH, HID);
  k_edge_agg256<<<cdiv((long)E * 64, TB), TB, 0, stream>>>(src, dst, dinv, bufA, bufB, E);
  k_bias_relu<<<cdiv(NH, TB), TB, 0, stream>>>(bufB, b1, bufA, HID, NH);

  // Layer 2 (agg init kept separate: fusion would alias GEMM input bufA)
  k_gemm_wmma<<<gGemm, TB, 0, stream>>>(bufA, W2, bufB, nullptr, dinv, HID, HID);
  k_scale_self<<<cdiv(NH, TB), TB, 0, stream>>>(bufB, dinv, bufA, HID, NH);
  k_edge_agg256<<<cdiv((long)E * 64, TB), TB, 0, stream>>>(src, dst, dinv, bufB, bufA, E);
  k_bias_relu<<<cdiv(NH, TB), TB, 0, stream>>>(bufA, b2, bufB, HID, NH);

  // Layer 3: hw3 = h2 @ W3 (+ fused self-loop init of out); out += edges; out += b3
  k_gemm_small<<<cdiv(NO, TB), TB, 0, stream>>>(bufB, W3, hw3, out, dinv, HID, OUTC, NO);
  k_edge_agg2<<<cdiv(E, TB), TB, 0, stream>>>(src, dst, dinv, hw3, out, E);
  k_bias<<<cdiv(NO, TB), TB, 0, stream>>>(out, b3, OUTC, NO);
}